// OffroadLossCenterlines_24670292148304
// MI455X (gfx1250) — compile-verified
//
#include <hip/hip_runtime.h>

typedef __attribute__((ext_vector_type(2))) float v2f;
typedef __attribute__((ext_vector_type(8))) float v8f;

#define THRESHOLD 0.5f
#define EPSF 1e-6f

// Shapes: points (32,6,64,2)  road_boundary (32,8,128,2)  centerlines (32,32,64,2)
// Output: (32,6) float32
__launch_bounds__(128)
__global__ void offroad_loss_kernel(const float* __restrict__ points,
                                    const float* __restrict__ boundary,
                                    const float* __restrict__ cls,
                                    float* __restrict__ out)
{
    constexpr int Cc = 6, T = 64, K = 8, N = 128, SP = 2048;

    __shared__ float sbx[K * N], sby[K * N];   // boundary points, 8 KB
    __shared__ float scx[SP],   scy[SP];       // centerline points, 16 KB
    __shared__ float spx[T],    spy[T];        // this (b,c)'s points
    __shared__ float ccx[T],    ccy[T];        // closest centerline pt per point
    __shared__ float smin[128];
    __shared__ int   sflag[128];
    __shared__ float sred[64];

    const int tid = threadIdx.x;
    const int blk = blockIdx.x;          // 0..191
    const int b   = blk / Cc;
    const int c   = blk % Cc;

    // ---------------- stage 0: stage inputs into LDS ----------------
    const float* bptr = boundary + (size_t)b * K * N * 2;
    for (int i = tid; i < K * N; i += 128) { sbx[i] = bptr[2 * i]; sby[i] = bptr[2 * i + 1]; }
    const float* cptr = cls + (size_t)b * SP * 2;
    for (int i = tid; i < SP; i += 128)    { scx[i] = cptr[2 * i]; scy[i] = cptr[2 * i + 1]; }
    const float* pptr = points + ((size_t)b * Cc + c) * T * 2;
    if (tid < T) { spx[tid] = pptr[2 * tid]; spy[tid] = pptr[2 * tid + 1]; }
    __syncthreads();

    const int t    = tid & 63;     // point index handled for stages 1/3
    const int half = tid >> 6;     // 2 threads per point split the segments
    const float px = spx[t], py = spy[t];

    // ---------------- stage 1: min squared distance to boundary segments ----------------
    float md2 = 3.4e38f;
    for (int k = 0; k < K; ++k) {
        const int base = k * N;
        for (int j = half; j < N - 1; j += 2) {
            float ax = sbx[base + j],     ay = sby[base + j];
            float dx = sbx[base + j + 1] - ax, dy = sby[base + j + 1] - ay;
            float v1x = px - ax, v1y = py - ay;
            float dt  = v1x * dx + v1y * dy;
            float esq = dx * dx + dy * dy;
            float proj = dt / (esq + EPSF);
            proj = fminf(fmaxf(proj, 0.0f), 1.0f);
            float qx = ax + dx * proj, qy = ay + dy * proj;
            float ex = px - qx, ey = py - qy;
            md2 = fminf(md2, ex * ex + ey * ey);
        }
    }
    smin[tid] = md2;

    // ---------------- stage 2: nearest centerline point via V_WMMA_F32_16X16X4_F32 ----------------
    // argmin_n |p-c|^2  ==  argmin_n (|c|^2 - 2 p.c);  p.c computed as 16x16x4 f32 WMMA (K=2 padded to 4).
    const int lane = tid & 31;
    const int w    = tid >> 5;        // wave id 0..3 -> M-tile (16 points each)
    const int nl   = lane & 15;

    // A fragment: lanes 0-15 hold K=0,1 of row M=w*16+nl; lanes 16-31 hold K=2,3 == 0.
    const int mrow = w * 16 + nl;
    v2f afrag;
    afrag.x = (lane < 16) ? spx[mrow] : 0.0f;
    afrag.y = (lane < 16) ? spy[mrow] : 0.0f;

    float best[8];
    int   bidx[8];
#pragma unroll
    for (int r = 0; r < 8; ++r) { best[r] = 3.4e38f; bidx[r] = 0; }

    for (int nt = 0; nt < SP / 16; ++nt) {
        const int n = nt * 16 + nl;
        float cx = scx[n], cy = scy[n];
        // B fragment: lanes 0-15 hold K=0,1 of column N=nl; lanes 16-31 hold K=2,3 == 0.
        v2f bfrag;
        bfrag.x = (lane < 16) ? cx : 0.0f;
        bfrag.y = (lane < 16) ? cy : 0.0f;
        v8f acc = {};
        acc = __builtin_amdgcn_wmma_f32_16x16x4_f32(false, afrag, false, bfrag,
                                                    (short)0, acc, false, false);
        float cn = cx * cx + cy * cy;
#pragma unroll
        for (int r = 0; r < 8; ++r) {
            // VGPR r: lanes 0-15 -> row r, lanes 16-31 -> row r+8; column = lane&15.
            // Strict < suffices for argmin's first-index tie rule within a lane,
            // because n is strictly increasing across iterations.
            float score  = cn - 2.0f * acc[r];
            bool  better = (score < best[r]);
            best[r] = better ? score : best[r];
            bidx[r] = better ? n     : bidx[r];
        }
    }
    // Reduce min(+argmin) across each 16-lane group; group 0 -> row r, group 1 -> row r+8.
    // Here indices arrive in arbitrary lane order, so ties break to the lower index.
#pragma unroll
    for (int r = 0; r < 8; ++r) {
        float v  = best[r];
        int   ix = bidx[r];
#pragma unroll
        for (int off = 8; off >= 1; off >>= 1) {
            float ov = __shfl_xor(v,  off, 16);
            int   oi = __shfl_xor(ix, off, 16);
            bool take = (ov < v) || (ov == v && oi < ix);
            v  = take ? ov : v;
            ix = take ? oi : ix;
        }
        if (nl == 0) {
            int m = w * 16 + r + ((lane >> 4) ? 8 : 0);
            ccx[m] = scx[ix];
            ccy[m] = scy[ix];
        }
    }
    __syncthreads();

    // ---------------- stage 3: intersection test (point -> closest_cl) vs boundary segments ----------------
    const float dax = ccx[t] - px, day = ccy[t] - py;   // da = a2 - a1
    int hit = 0;
    for (int k = 0; k < K; ++k) {
        const int base = k * N;
        for (int j = half; j < N - 1; j += 2) {
            float b1x = sbx[base + j],     b1y = sby[base + j];
            float dbx = sbx[base + j + 1] - b1x, dby = sby[base + j + 1] - b1y;
            float dpx = px - b1x, dpy = py - b1y;        // dp = a1 - b1
            float dadb = dax * dby - day * dbx;
            float inv  = 1.0f / (dadb + EPSF);
            float tt = (dax * dpy - day * dpx) * inv;
            float uu = (dbx * dpy - dby * dpx) * inv;
            hit |= (tt >= 0.0f) & (tt <= 1.0f) & (uu >= 0.0f) & (uu <= 1.0f);
        }
    }
    sflag[tid] = hit;
    __syncthreads();

    // ---------------- stage 4: sign, relu, sum over T ----------------
    if (tid < 64) {
        float mind   = sqrtf(fminf(smin[tid], smin[tid + 64]));
        int   anyhit = sflag[tid] | sflag[tid + 64];
        float sd     = anyhit ? mind : -mind;            // inside (no hit) -> negative
        sred[tid]    = fmaxf(sd + THRESHOLD, 0.0f);
    }
    __syncthreads();
    for (int s = 32; s >= 1; s >>= 1) {
        if (tid < s) sred[tid] += sred[tid + s];
        __syncthreads();
    }
    if (tid == 0) out[blk] = sred[0];
}

extern "C" void kernel_launch(void* const* d_in, const int* in_sizes, int n_in,
                              void* d_out, int out_size, void* d_ws, size_t ws_size,
                              hipStream_t stream) {
    (void)in_sizes; (void)n_in; (void)out_size; (void)d_ws; (void)ws_size;
    const float* points   = (const float*)d_in[0];   // (32,6,64,2)
    const float* boundary = (const float*)d_in[1];   // (32,8,128,2)
    const float* cls      = (const float*)d_in[2];   // (32,32,64,2)
    float* out            = (float*)d_out;           // (32,6)
    offroad_loss_kernel<<<dim3(32 * 6), dim3(128), 0, stream>>>(points, boundary, cls, out);
}